// TSPGNN_29652454211949
// MI455X (gfx1250) — compile-verified
//
#include <hip/hip_runtime.h>
#include <cmath>

typedef __attribute__((ext_vector_type(2))) float v2f;
typedef __attribute__((ext_vector_type(8))) float v8f;

// Native no-return fp32 atomic add (STOREcnt-tracked; S_ENDPGM's implicit
// wait-idle drains it). Device scope for cross-workgroup coherence.
static __device__ __forceinline__ void atomAddF(float* p, float v) {
    asm volatile("global_atomic_add_f32 %0, %1, off scope:SCOPE_DEV"
                 :: "v"(p), "v"(v) : "memory");
}

// ---------------- degree / normalization ----------------
__global__ void k_deg_init(float* deg, int N) {
    int v = blockIdx.x * blockDim.x + threadIdx.x;
    if (v < N) deg[v] = 1.0f;  // self-loop
}

__global__ void k_deg_count(const int* ei, float* deg, int E) {
    int e = blockIdx.x * blockDim.x + threadIdx.x;
    if (e < E) atomAddF(&deg[ei[E + e]], 1.0f);  // col = second row of edge_index
}

__global__ void k_dinv(const float* deg, float* dinv, int N) {
    int v = blockIdx.x * blockDim.x + threadIdx.x;
    if (v < N) dinv[v] = rsqrtf(deg[v]);  // deg >= 1 always
}

// ---------------- layer 1: x[N,2] @ W1[2,64], self-loop init ----------------
__global__ void k_transform1(const float* x, const float* W1, const float* dinv,
                             float* t1, float* agg1, int N) {
    int v = blockIdx.x * blockDim.x + threadIdx.x;
    if (v >= N) return;
    float x0 = x[2 * v], x1 = x[2 * v + 1];
    float s = dinv[v] * dinv[v];
    const float4* w0 = (const float4*)W1;
    const float4* w1 = (const float4*)(W1 + 64);
    float4* t1p = (float4*)(t1 + v * 64);
    float4* a1p = (float4*)(agg1 + v * 64);
#pragma unroll
    for (int j = 0; j < 16; ++j) {
        float4 a = w0[j], b = w1[j], t, g;
        t.x = x0 * a.x + x1 * b.x; t.y = x0 * a.y + x1 * b.y;
        t.z = x0 * a.z + x1 * b.z; t.w = x0 * a.w + x1 * b.w;
        t1p[j] = t;
        g.x = t.x * s; g.y = t.y * s; g.z = t.z * s; g.w = t.w * s;
        a1p[j] = g;
    }
}

__global__ void k_scatter1(const int* ei, const float* dinv, const float* t1,
                           float* agg1, int E) {
    int tid = blockIdx.x * blockDim.x + threadIdx.x;
    if (tid >= 4 * E) return;
    int e = tid >> 2, q = tid & 3;
    int r = ei[e], c = ei[E + e];
    float norm = dinv[r] * dinv[c];
    const float4* src = (const float4*)(t1 + r * 64 + q * 16);
    float* dst = agg1 + c * 64 + q * 16;
#pragma unroll
    for (int i = 0; i < 4; ++i) {
        float4 m = src[i];
        atomAddF(dst + 4 * i + 0, m.x * norm);
        atomAddF(dst + 4 * i + 1, m.y * norm);
        atomAddF(dst + 4 * i + 2, m.z * norm);
        atomAddF(dst + 4 * i + 3, m.w * norm);
    }
}

__global__ void k_finalize1(float* agg1, const float* b1, int N) {
    int tid = blockIdx.x * blockDim.x + threadIdx.x;
    int base = tid * 4;
    if (base >= N * 64) return;
    float4 a = *(float4*)(agg1 + base);
    float4 b = *(const float4*)(b1 + (base & 63));
    a.x = fmaxf(a.x + b.x, 0.f); a.y = fmaxf(a.y + b.y, 0.f);
    a.z = fmaxf(a.z + b.z, 0.f); a.w = fmaxf(a.w + b.w, 0.f);
    *(float4*)(agg1 + base) = a;  // in place -> h1
}

// ---------------- layer 2 transform: h1[N,64] @ W2[64,32] via f32 WMMA ----------------
__global__ __launch_bounds__(256) void k_transform2(const float* h1, const float* W2,
                                                    float* t2, int N) {
    __shared__ float sA[8 * 1024];  // 8 waves x 16 nodes x 64 feats
    int tid = threadIdx.x, wave = tid >> 5, lane = tid & 31;
    int nb = blockIdx.x * 128 + wave * 16;
    int part = lane >> 4, m = lane & 15;
    int node = nb + m;
    float* dst = sA + wave * 1024 + m * 64 + part * 32;
    if (node < N) {
        const float4* src = (const float4*)(h1 + node * 64 + part * 32);
#pragma unroll
        for (int i = 0; i < 8; ++i) ((float4*)dst)[i] = src[i];
    } else {
        float4 z = {0.f, 0.f, 0.f, 0.f};
#pragma unroll
        for (int i = 0; i < 8; ++i) ((float4*)dst)[i] = z;
    }
    __syncthreads();

    int hi = lane >> 4, n = lane & 15;
    v8f acc0 = {}; v8f acc1 = {};
    const float* aBase = sA + wave * 1024 + (lane & 15) * 64;  // A: M = lane&15
#pragma unroll
    for (int k = 0; k < 16; ++k) {
        // A frag (16x4 f32): lanes 0-15 hold K = 4k+{0,1}; lanes 16-31 hold K = 4k+{2,3}
        v2f a = *(const v2f*)(aBase + 4 * k + 2 * hi);
        int r0 = 4 * k + 2 * hi;
        v2f b0, b1;
        b0.x = W2[r0 * 32 + n];          b0.y = W2[(r0 + 1) * 32 + n];
        b1.x = W2[r0 * 32 + 16 + n];     b1.y = W2[(r0 + 1) * 32 + 16 + n];
        acc0 = __builtin_amdgcn_wmma_f32_16x16x4_f32(false, a, false, b0, (short)0, acc0, false, false);
        acc1 = __builtin_amdgcn_wmma_f32_16x16x4_f32(false, a, false, b1, (short)0, acc1, false, false);
    }
#pragma unroll
    for (int v = 0; v < 8; ++v) {
        int node2 = nb + v + 8 * hi;  // D: M = v + 8*(lane>=16), N = lane&15
        if (node2 < N) {
            t2[node2 * 32 + n]      = acc0[v];
            t2[node2 * 32 + 16 + n] = acc1[v];
        }
    }
}

__global__ void k_agg2init(const float* t2, const float* dinv, float* agg2, int N) {
    int tid = blockIdx.x * blockDim.x + threadIdx.x;
    int base = tid * 4;
    if (base >= N * 32) return;
    int v = base >> 5;
    float s = dinv[v] * dinv[v];
    float4 t = *(const float4*)(t2 + base);
    t.x *= s; t.y *= s; t.z *= s; t.w *= s;
    *(float4*)(agg2 + base) = t;
}

__global__ void k_scatter2(const int* ei, const float* dinv, const float* t2,
                           float* agg2, int E) {
    int tid = blockIdx.x * blockDim.x + threadIdx.x;
    if (tid >= 4 * E) return;
    int e = tid >> 2, q = tid & 3;
    int r = ei[e], c = ei[E + e];
    float norm = dinv[r] * dinv[c];
    const float4* src = (const float4*)(t2 + r * 32 + q * 8);
    float* dst = agg2 + c * 32 + q * 8;
#pragma unroll
    for (int i = 0; i < 2; ++i) {
        float4 m = src[i];
        atomAddF(dst + 4 * i + 0, m.x * norm);
        atomAddF(dst + 4 * i + 1, m.y * norm);
        atomAddF(dst + 4 * i + 2, m.z * norm);
        atomAddF(dst + 4 * i + 3, m.w * norm);
    }
}

__global__ void k_finalize2(float* agg2, const float* b2, int N) {
    int tid = blockIdx.x * blockDim.x + threadIdx.x;
    int base = tid * 4;
    if (base >= N * 32) return;
    float4 a = *(float4*)(agg2 + base);
    float4 b = *(const float4*)(b2 + (base & 31));
    a.x = fmaxf(a.x + b.x, 0.f); a.y = fmaxf(a.y + b.y, 0.f);
    a.z = fmaxf(a.z + b.z, 0.f); a.w = fmaxf(a.w + b.w, 0.f);
    *(float4*)(agg2 + base) = a;  // in place -> h2
}

// ---------------- edge MLP: ef[16e,65] @ mW1[65,16] via f32 WMMA, then 16->1 ----------------
__global__ __launch_bounds__(256) void k_edge_mlp(const int* ei, const float* h2,
                                                  const float* ea, const float* mW1,
                                                  const float* mb1, const float* mW2,
                                                  const float* mb2, float* out, int E) {
    __shared__ float sE[8 * 1088];  // 8 waves x 16 edges x 68 (65 feats + pad)
    __shared__ float sH[8 * 256];   // 8 waves x 16 edges x 16 hidden
    int tid = threadIdx.x, wave = tid >> 5, lane = tid & 31;
    int eb = blockIdx.x * 128;
    int part = lane >> 4, m = lane & 15;
    int e = eb + wave * 16 + m;
    float* dst = sE + wave * 1088 + m * 68 + part * 32;
    if (e < E) {
        int idx = part ? ei[E + e] : ei[e];  // part0: h2[row] -> cols 0..31, part1: h2[col] -> 32..63
        const float4* src = (const float4*)(h2 + idx * 32);
#pragma unroll
        for (int i = 0; i < 8; ++i) ((float4*)dst)[i] = src[i];
        if (part) {
            float4 pad = {ea[e], 0.f, 0.f, 0.f};  // col 64 = edge_attr, 65..67 = 0
            *(float4*)(sE + wave * 1088 + m * 68 + 64) = pad;
        }
    } else {
        float4 z = {0.f, 0.f, 0.f, 0.f};
#pragma unroll
        for (int i = 0; i < 8; ++i) ((float4*)dst)[i] = z;
        if (part) *(float4*)(sE + wave * 1088 + m * 68 + 64) = z;
    }
    __syncthreads();

    int hi = lane >> 4, n = lane & 15;
    float bn = mb1[n];
    v8f acc = {};
    const float* aBase = sE + wave * 1088 + (lane & 15) * 68;
#pragma unroll
    for (int k = 0; k < 17; ++k) {  // K = 65 padded to 68 = 17 x 4
        v2f a = *(const v2f*)(aBase + 4 * k + 2 * hi);
        int r0 = 4 * k + 2 * hi;
        v2f b;
        b.x = (r0 < 65)     ? mW1[r0 * 16 + n]       : 0.0f;
        b.y = (r0 + 1 < 65) ? mW1[(r0 + 1) * 16 + n] : 0.0f;
        acc = __builtin_amdgcn_wmma_f32_16x16x4_f32(false, a, false, b, (short)0, acc, false, false);
    }
#pragma unroll
    for (int v = 0; v < 8; ++v) {
        int mo = v + 8 * hi;
        sH[wave * 256 + mo * 16 + n] = fmaxf(acc[v] + bn, 0.f);
    }
    __syncthreads();

    if (tid < 128) {
        int e2 = eb + tid;
        if (e2 < E) {
            float s = 0.f;
#pragma unroll
            for (int j = 0; j < 16; ++j) s += sH[tid * 16 + j] * mW2[j];
            s += mb2[0];
            out[e2] = 1.0f / (1.0f + expf(-s));
        }
    }
}

// ---------------- launch ----------------
extern "C" void kernel_launch(void* const* d_in, const int* in_sizes, int n_in,
                              void* d_out, int out_size, void* d_ws, size_t ws_size,
                              hipStream_t stream) {
    (void)n_in; (void)out_size; (void)ws_size;
    const float* x   = (const float*)d_in[0];
    const int*   ei  = (const int*)d_in[1];
    const float* ea  = (const float*)d_in[2];
    const float* W1  = (const float*)d_in[4];
    const float* b1  = (const float*)d_in[5];
    const float* W2  = (const float*)d_in[6];
    const float* b2  = (const float*)d_in[7];
    const float* mW1 = (const float*)d_in[8];
    const float* mb1 = (const float*)d_in[9];
    const float* mW2 = (const float*)d_in[10];
    const float* mb2 = (const float*)d_in[11];
    float* out = (float*)d_out;

    int N = in_sizes[0] / 2;   // 100000
    int E = in_sizes[2];       // 1600000

    float* ws   = (float*)d_ws;
    float* dinv = ws;                           // N floats
    float* bufA = ws + N;                       // N*64: deg, then t1, then t2
    float* bufB = bufA + (size_t)N * 64;        // N*64: agg1/h1, then agg2/h2
    float* deg  = bufA;

    const int B = 256;
    // degree + norm
    k_deg_init <<<(N + B - 1) / B, B, 0, stream>>>(deg, N);
    k_deg_count<<<(E + B - 1) / B, B, 0, stream>>>(ei, deg, E);
    k_dinv     <<<(N + B - 1) / B, B, 0, stream>>>(deg, dinv, N);
    // layer 1
    k_transform1<<<(N + B - 1) / B, B, 0, stream>>>(x, W1, dinv, bufA, bufB, N);
    k_scatter1  <<<(4 * E + B - 1) / B, B, 0, stream>>>(ei, dinv, bufA, bufB, E);
    k_finalize1 <<<(N * 16 + B - 1) / B, B, 0, stream>>>(bufB, b1, N);     // bufB = h1
    // layer 2
    k_transform2<<<(N + 127) / 128, B, 0, stream>>>(bufB, W2, bufA, N);    // bufA = t2
    k_agg2init  <<<(N * 8 + B - 1) / B, B, 0, stream>>>(bufA, dinv, bufB, N);
    k_scatter2  <<<(4 * E + B - 1) / B, B, 0, stream>>>(ei, dinv, bufA, bufB, E);
    k_finalize2 <<<(N * 8 + B - 1) / B, B, 0, stream>>>(bufB, b2, N);      // bufB = h2
    // edge MLP
    k_edge_mlp  <<<(E + 127) / 128, B, 0, stream>>>(ei, bufB, ea, mW1, mb1, mW2, mb2, out, E);
}